// QuantumConvFilter_41377714929699
// MI455X (gfx1250) — compile-verified
//
#include <hip/hip_runtime.h>
#include <hip/hip_bf16.h>

// CDNA5 / gfx1250, wave32. State tile contract (matches WMMA f32 16x16 C/D
// layout, M = state dim, N = batch element):
//   lane L, VGPR r  holds  dim d = r + 8*(L>=16 ? 1:0)  of element n = L&15.

typedef __attribute__((ext_vector_type(2))) float v2f;
typedef __attribute__((ext_vector_type(8))) float v8f;

#define N_QUBITS 4
#define N_LAYERS 3
#define DIM 16

// ---------------------------------------------------------------------------
// Prelude: build per-layer fused unitaries U_l = P_cnot_ring * Prod_i RZ_i RY_i
// as 16x16 complex, stored row-major: Wbuf[layer][comp(0=re,1=im)][row][col].
// One thread per (layer, column): 48 active threads.
// ---------------------------------------------------------------------------
__global__ void qcf_build_unitaries(const float* __restrict__ w, float* __restrict__ Wbuf) {
  int t = threadIdx.x;
  if (t >= N_LAYERS * DIM) return;
  int layer = t / DIM, col = t % DIM;

  float re[DIM], im[DIM];
  for (int d = 0; d < DIM; ++d) { re[d] = 0.0f; im[d] = 0.0f; }
  re[col] = 1.0f;

  for (int i = 0; i < N_QUBITS; ++i) {
    // RY(w[layer][i][0])
    float th = 0.5f * w[(layer * N_QUBITS + i) * 3 + 0];
    float c = __cosf(th), s = __sinf(th);
    int stride = 1 << (3 - i);
    for (int d = 0; d < DIM; ++d) {
      if ((d & stride) == 0) {
        int d2 = d + stride;
        float ar = re[d], ai = im[d], br = re[d2], bi = im[d2];
        re[d]  = c * ar - s * br;  im[d]  = c * ai - s * bi;
        re[d2] = s * ar + c * br;  im[d2] = s * ai + c * bi;
      }
    }
    // RZ(w[layer][i][1]): bit==0 -> *exp(-i*ph), bit==1 -> *exp(+i*ph)
    float ph = 0.5f * w[(layer * N_QUBITS + i) * 3 + 1];
    float cp = __cosf(ph), sp = __sinf(ph);
    for (int d = 0; d < DIM; ++d) {
      float sgn = ((d >> (3 - i)) & 1) ? sp : -sp;
      float r0 = re[d], i0 = im[d];
      re[d] = cp * r0 - sgn * i0;
      im[d] = cp * i0 + sgn * r0;
    }
  }
  // CNOT ring (0,1),(1,2),(2,3),(3,0): new[j] = old[perm[j]]
  const int ct[4][2] = {{0, 1}, {1, 2}, {2, 3}, {3, 0}};
  for (int k = 0; k < 4; ++k) {
    int ctrl = ct[k][0], tgt = ct[k][1];
    float nr[DIM], ni[DIM];
    for (int j = 0; j < DIM; ++j) {
      int cbit = (j >> (3 - ctrl)) & 1;
      int src = cbit ? (j ^ (1 << (3 - tgt))) : j;
      nr[j] = re[src]; ni[j] = im[src];
    }
    for (int j = 0; j < DIM; ++j) { re[j] = nr[j]; im[j] = ni[j]; }
  }
  for (int row = 0; row < DIM; ++row) {
    Wbuf[((layer * 2 + 0) * DIM + row) * DIM + col] = re[row];
    Wbuf[((layer * 2 + 1) * DIM + row) * DIM + col] = im[row];
  }
}

// ---------------------------------------------------------------------------
// Main kernel helpers
// ---------------------------------------------------------------------------
__device__ inline v8f wmma4(v2f a, v2f b, v8f c) {
  // D(16x16,f32) = A(16x4,f32) x B(4x16,f32) + C
  return __builtin_amdgcn_wmma_f32_16x16x4_f32(
      /*neg_a=*/false, a, /*neg_b=*/false, b,
      /*c_mod=*/(short)0, c, /*reuse_a=*/false, /*reuse_b=*/false);
}

// RY on wire 0: pairs (d, d+8) live in opposite lane halves -> half-wave swap.
__device__ inline void ry_wire0(float (&sr)[8], float (&si)[8],
                                float c, float s, bool lo) {
  float ss = lo ? -s : s;   // lo lanes hold 'a': c*a - s*b ; hi hold 'b': s*a + c*b
#pragma unroll
  for (int r = 0; r < 8; ++r) {
    float pr = __shfl_xor(sr[r], 16, 32);
    float pi = __shfl_xor(si[r], 16, 32);
    sr[r] = c * sr[r] + ss * pr;
    si[r] = c * si[r] + ss * pi;
  }
}

// RY on wires 1..3: pairs within the lane-local 8 dims (stride 4/2/1).
template <int STRIDE>
__device__ inline void ry_local(float (&sr)[8], float (&si)[8], float c, float s) {
#pragma unroll
  for (int r = 0; r < 8; ++r) {
    if ((r & STRIDE) == 0) {
      int r2 = r + STRIDE;
      float ar = sr[r], ai = si[r], br = sr[r2], bi = si[r2];
      sr[r]  = c * ar - s * br;  si[r]  = c * ai - s * bi;
      sr[r2] = s * ar + c * br;  si[r2] = s * ai + c * bi;
    }
  }
}

// Repack state (C/D layout) into 4 B-matrix K-slices (4x16 f32 each).
// Assumed B layout: VGPR0 = row K0 (lanes 0-15) | row K0+2 (lanes 16-31),
//                   VGPR1 = row K0+1 | K0+3.
__device__ inline void build_b_slices(const float (&s)[8], bool lo, v2f (&b)[4]) {
  float t0 = __shfl_xor(s[0], 16, 32), t1 = __shfl_xor(s[1], 16, 32);
  float t2 = __shfl_xor(s[2], 16, 32), t3 = __shfl_xor(s[3], 16, 32);
  float t4 = __shfl_xor(s[4], 16, 32), t5 = __shfl_xor(s[5], 16, 32);
  float t6 = __shfl_xor(s[6], 16, 32), t7 = __shfl_xor(s[7], 16, 32);
  b[0].x = lo ? s[0] : t2;  b[0].y = lo ? s[1] : t3;   // dims 0..3
  b[1].x = lo ? s[4] : t6;  b[1].y = lo ? s[5] : t7;   // dims 4..7
  b[2].x = lo ? t0 : s[2];  b[2].y = lo ? t1 : s[3];   // dims 8..11
  b[3].x = lo ? t4 : s[6];  b[3].y = lo ? t5 : s[7];   // dims 12..15
}

// ---------------------------------------------------------------------------
// Main kernel: one wave32 processes 16 batch elements.
// ---------------------------------------------------------------------------
__global__ __launch_bounds__(256) void qcf_main(const float* __restrict__ x,
                                                const float* __restrict__ w,
                                                const float* __restrict__ Wbuf,
                                                float* __restrict__ out, int B) {
  const int lane = threadIdx.x & 31;
  const bool lo = lane < 16;
  const int hi = lo ? 0 : 1;
  const int n = lane & 15;
  const int wave = (int)((blockIdx.x * blockDim.x + threadIdx.x) >> 5);
  const int e0 = wave * 16;
  const int elem = e0 + n;
  const bool valid = elem < B;
  const int eclamp = valid ? elem : (B - 1);

  // x for this lane's element (lanes n and n+16 share element n)
  float4 xv4 = reinterpret_cast<const float4*>(x)[eclamp];
  float xa[4] = {xv4.x, xv4.y, xv4.z, xv4.w};

  // |psi0> = RY(x)|0> = product state; bit_i(d) = (d >> (3-i)) & 1
  float fc[4], fs[4];
#pragma unroll
  for (int i = 0; i < 4; ++i) __sincosf(0.5f * xa[i], &fs[i], &fc[i]);

  float sr[8], si[8];
#pragma unroll
  for (int r = 0; r < 8; ++r) {
    int d = r + 8 * hi;
    float v = (((d >> 3) & 1) ? fs[0] : fc[0]) *
              (((d >> 2) & 1) ? fs[1] : fc[1]) *
              (((d >> 1) & 1) ? fs[2] : fc[2]) *
              (((d     ) & 1) ? fs[3] : fc[3]);
    sr[r] = v;
    si[r] = 0.0f;
  }

#pragma unroll
  for (int layer = 0; layer < N_LAYERS; ++layer) {
    // ---- fused weight unitary via WMMA: Y = U_l * Psi (complex) ----
    const float* Wr = Wbuf + (layer * 2 + 0) * 256;
    const float* Wi = Wbuf + (layer * 2 + 1) * 256;
    // A-slice layout (16x4 f32): lane M = lane&15; VGPR0 K = base+2*hi, VGPR1 K+1
    v2f ar[4], ai[4];
#pragma unroll
    for (int j = 0; j < 4; ++j) {
      int off = n * 16 + 4 * j + 2 * hi;
      ar[j] = *reinterpret_cast<const v2f*>(Wr + off);
      ai[j] = *reinterpret_cast<const v2f*>(Wi + off);
    }
    v2f br[4], bi[4], bni[4];
    build_b_slices(sr, lo, br);
    build_b_slices(si, lo, bi);
#pragma unroll
    for (int j = 0; j < 4; ++j) { bni[j].x = -bi[j].x; bni[j].y = -bi[j].y; }

    v8f yr = {}; v8f yi = {};
#pragma unroll
    for (int j = 0; j < 4; ++j) {
      yr = wmma4(ar[j], br[j], yr);    // + Wr*Pr
      yi = wmma4(ar[j], bi[j], yi);    // + Wr*Pi
    }
#pragma unroll
    for (int j = 0; j < 4; ++j) {
      yr = wmma4(ai[j], bni[j], yr);   // - Wi*Pi
      yi = wmma4(ai[j], br[j], yi);    // + Wi*Pr
    }
#pragma unroll
    for (int r = 0; r < 8; ++r) { sr[r] = yr[r]; si[r] = yi[r]; }

    // ---- per-element RY(x_i * w[layer,i,2]) ----
#pragma unroll
    for (int i = 0; i < 4; ++i) {
      float th = 0.5f * xa[i] * w[(layer * N_QUBITS + i) * 3 + 2];
      float c, s;
      __sincosf(th, &s, &c);
      if (i == 0)      ry_wire0(sr, si, c, s, lo);
      else if (i == 1) ry_local<4>(sr, si, c, s);
      else if (i == 2) ry_local<2>(sr, si, c, s);
      else             ry_local<1>(sr, si, c, s);
    }
  }

  // ---- Z expectations: e[i] = sum_d (1-2*bit_i(d)) * |psi_d|^2 ----
  float e[4] = {0.0f, 0.0f, 0.0f, 0.0f};
#pragma unroll
  for (int r = 0; r < 8; ++r) {
    int d = r + 8 * hi;
    float p = sr[r] * sr[r] + si[r] * si[r];
#pragma unroll
    for (int i = 0; i < 4; ++i) e[i] += (((d >> (3 - i)) & 1) ? -p : p);
  }
#pragma unroll
  for (int i = 0; i < 4; ++i) e[i] += __shfl_xor(e[i], 16, 32);

  if (lo && valid) {
    float4 o; o.x = e[0]; o.y = e[1]; o.z = e[2]; o.w = e[3];
    reinterpret_cast<float4*>(out)[elem] = o;
  }
}

// ---------------------------------------------------------------------------
extern "C" void kernel_launch(void* const* d_in, const int* in_sizes, int n_in,
                              void* d_out, int out_size, void* d_ws, size_t ws_size,
                              hipStream_t stream) {
  const float* x = (const float*)d_in[0];       // (B, 4) f32
  const float* w = (const float*)d_in[1];       // (3, 4, 3) f32
  float* out = (float*)d_out;                   // (B, 4) f32
  float* Wbuf = (float*)d_ws;                   // 3*2*16*16 f32 = 6 KB

  int B = in_sizes[0] / N_QUBITS;

  qcf_build_unitaries<<<1, 64, 0, stream>>>(w, Wbuf);

  int waves = (B + 15) / 16;
  int threads = 256;                            // 8 waves / block
  int blocks = (waves * 32 + threads - 1) / threads;
  qcf_main<<<blocks, threads, 0, stream>>>(x, w, Wbuf, out, B);
}